// DWT_30382598652218
// MI455X (gfx1250) — compile-verified
//
#include <hip/hip_runtime.h>

typedef __attribute__((ext_vector_type(2))) float v2f;
typedef __attribute__((ext_vector_type(8))) float v8f;

// db4 decomposition filters, pre-reversed (correlation form used by the reference):
// KL[j] = DEC_LO[7-j], KH[j] = DEC_HI[7-j]
// out_lo[i] = sum_j xp[2i+j]*KL[j], xp = x padded with 6 zeros on the left.

__global__ __launch_bounds__(256) void dwt_db4_level_wmma(
    const float* __restrict__ in,   // [64, Nin]
    float*       __restrict__ cA,   // [64, Mout]
    float*       __restrict__ cD,   // [64, Mout]
    int Nin, int Mout, int tilesPerRow, int totalWaves)
{
    constexpr float KL[8] = {
         0.23037781330885523f,  0.7148465705525415f,  0.6308807679295904f,
        -0.02798376941698385f, -0.18703481171888114f, 0.030841381835986965f,
         0.032883011666982945f, -0.010597401784997278f };
    constexpr float KH[8] = {
        -0.010597401784997278f, -0.032883011666982945f, 0.030841381835986965f,
         0.18703481171888114f,  -0.02798376941698385f,  -0.6308807679295904f,
         0.7148465705525415f,   -0.23037781330885523f };

    const int tid  = threadIdx.x;
    const int lane = tid & 31;
    const int w    = blockIdx.x * 8 + (tid >> 5);
    if (w >= totalWaves) return;          // uniform per wave -> EXEC stays all-ones

    const int row  = w / tilesPerRow;
    const int tile = w - row * tilesPerRow;
    const int base = tile * 16;

    const int n  = lane & 15;             // column within tile (output index / filter row)
    const int kh = lane >> 4;             // 0: K rows 0,1 ; 1: K rows 2,3
    const size_t inRow = (size_t)row * (size_t)Nin;

    const int ob = base + n;              // output element this column produces
    const int p0 = 2 * ob + 2 * kh - 6;   // position in original x of xp[2*ob + 2*kh]

    // ---- B operands: unconditional loads at double-clamped (always legal)
    //      addresses; zero-pad boundary realized with cndmask selects.  ----
    float bv[4];
    bool  bin[4];
#pragma unroll
    for (int t = 0; t < 4; ++t) {
        const int p  = p0 + (t >> 1) * 4 + (t & 1);      // taps p0, p0+1, p0+4, p0+5
        int pc = p;
        pc = pc < 0 ? 0 : pc;
        pc = pc >= Nin ? (Nin - 1) : pc;                 // both-side clamp: safe address
        bv[t]  = in[inRow + (size_t)pc];                 // unconditional coalesced load
        bin[t] = (p >= 0) && (p < Nin);
    }

    v8f c = {0.f,0.f,0.f,0.f,0.f,0.f,0.f,0.f};

#pragma unroll
    for (int g = 0; g < 2; ++g) {         // two K=4 chunks cover the 8 taps
        // ---- A operand: constant filter matrix (rows 0..1 nonzero) ----
        const float lo0 = kh ? KL[4*g+2] : KL[4*g+0];
        const float lo1 = kh ? KL[4*g+3] : KL[4*g+1];
        const float hi0 = kh ? KH[4*g+2] : KH[4*g+0];
        const float hi1 = kh ? KH[4*g+3] : KH[4*g+1];
        v2f A;
        A[0] = (n == 0) ? lo0 : ((n == 1) ? hi0 : 0.0f);
        A[1] = (n == 0) ? lo1 : ((n == 1) ? hi1 : 0.0f);

        v2f Bm;
        Bm[0] = bin[2*g + 0] ? bv[2*g + 0] : 0.0f;
        Bm[1] = bin[2*g + 1] ? bv[2*g + 1] : 0.0f;

        // D = A(16x4) x B(4x16) + C  ->  v_wmma_f32_16x16x4_f32
        c = __builtin_amdgcn_wmma_f32_16x16x4_f32(
                /*neg_a=*/false, A, /*neg_b=*/false, Bm,
                /*c_mod=*/(short)0, c, /*reuse_a=*/false, /*reuse_b=*/false);
    }

    // D layout: lanes 0..15, VGPR m = D[M=m][N=lane]; M=0 -> cA, M=1 -> cD.
    if (lane < 16 && ob < Mout) {
        const size_t outRow = (size_t)row * (size_t)Mout;
        cA[outRow + ob] = c[0];
        cD[outRow + ob] = c[1];
    }
}

extern "C" void kernel_launch(void* const* d_in, const int* in_sizes, int n_in,
                              void* d_out, int out_size, void* d_ws, size_t ws_size,
                              hipStream_t stream) {
    (void)in_sizes; (void)n_in; (void)out_size; (void)ws_size;

    const float* x  = (const float*)d_in[0];
    float* out      = (float*)d_out;
    float* ws       = (float*)d_ws;

    const int BATCH = 64;
    const int LEVEL = 6;

    // Level lengths: Nlev[0]=input N, Nlev[l]=length after level l
    int Nlev[LEVEL + 1];
    Nlev[0] = 262144;
    for (int l = 1; l <= LEVEL; ++l) Nlev[l] = (Nlev[l - 1] + 7) / 2;

    // Output layout: a6, d6, d5, d4, d3, d2, d1
    size_t offD[LEVEL + 1];
    size_t pos = (size_t)BATCH * (size_t)Nlev[LEVEL];   // after a6
    for (int l = LEVEL; l >= 1; --l) {
        offD[l] = pos;
        pos += (size_t)BATCH * (size_t)Nlev[l];
    }

    // Workspace ping-pong for approximation coefficients:
    // wsA holds cA of levels 1,3,5 (max 64*131075 floats = 33.6 MB),
    // wsB levels 2,4 (16.8 MB).
    float* wsA = ws;
    float* wsB = ws + (size_t)BATCH * (size_t)Nlev[1];

    const float* curIn = x;
    for (int l = 1; l <= LEVEL; ++l) {
        const int Nin  = Nlev[l - 1];
        const int Mout = Nlev[l];
        float* cA = (l == LEVEL) ? out : ((l & 1) ? wsA : wsB);
        float* cD = out + offD[l];

        const int tilesPerRow = (Mout + 15) / 16;
        const int totalWaves  = BATCH * tilesPerRow;
        const int blocks      = (totalWaves + 7) / 8;   // 8 wave32 per 256-thread block

        dwt_db4_level_wmma<<<blocks, 256, 0, stream>>>(
            curIn, cA, cD, Nin, Mout, tilesPerRow, totalWaves);

        curIn = cA;
    }
}